// MeanInterpolator_78683800862820
// MI455X (gfx1250) — compile-verified
//
#include <hip/hip_runtime.h>
#include <math.h>

// Problem constants (from reference)
#define X_DIM  8
#define H_DIM  128
#define H2_DIM 64
#define Z_DIM  16
#define M_CTX  2048
#define N_TGT  8192

typedef __attribute__((ext_vector_type(2))) float v2f;
typedef __attribute__((ext_vector_type(4))) float v4f;
typedef __attribute__((ext_vector_type(8))) float v8f;

// ---------------------------------------------------------------------------
// Stage 1: MLP features for context+target rows.
//   context rows  -> zcp[m,:] = z_c[m,:] @ (W + W^T),  c[m] = 0.5 * z_c . zcp
//   target rows   -> zt[n,:]  = z_t[n,:]
// One thread per row; all weights staged in LDS (~42 KB).
// ---------------------------------------------------------------------------
__global__ __launch_bounds__(256) void np_feat_kernel(
    const float* __restrict__ x_context, const float* __restrict__ x_target,
    const float* __restrict__ W1, const float* __restrict__ b1,
    const float* __restrict__ W2, const float* __restrict__ b2,
    const float* __restrict__ W3, const float* __restrict__ b3,
    const float* __restrict__ Wq,
    float* __restrict__ zcp_out, float* __restrict__ c_out,
    float* __restrict__ zt_out)
{
    __shared__ float sW1[X_DIM * H_DIM];
    __shared__ float sB1[H_DIM];
    __shared__ float sW2[H_DIM * H2_DIM];
    __shared__ float sB2[H2_DIM];
    __shared__ float sW3[H2_DIM * Z_DIM];
    __shared__ float sB3[Z_DIM];
    __shared__ float sS[Z_DIM * Z_DIM];   // S = W + W^T

    const int tid = threadIdx.x;
    for (int i = tid; i < X_DIM * H_DIM;  i += 256) sW1[i] = W1[i];
    for (int i = tid; i < H_DIM * H2_DIM; i += 256) sW2[i] = W2[i];
    for (int i = tid; i < H2_DIM * Z_DIM; i += 256) sW3[i] = W3[i];
    if (tid < H_DIM)  sB1[tid] = b1[tid];
    if (tid < H2_DIM) sB2[tid] = b2[tid];
    if (tid < Z_DIM)  sB3[tid] = b3[tid];
    {   // 16x16 symmetrized bilinear form
        int d = tid >> 4, e = tid & 15;
        if (tid < 256) sS[tid] = Wq[d * 16 + e] + Wq[e * 16 + d];
    }
    __syncthreads();

    const int row = blockIdx.x * 256 + tid;
    if (row >= M_CTX + N_TGT) return;
    const bool is_ctx = row < M_CTX;
    const float* xr = is_ctx ? (x_context + (size_t)row * X_DIM)
                             : (x_target + (size_t)(row - M_CTX) * X_DIM);
    float x8[X_DIM];
#pragma unroll
    for (int k = 0; k < X_DIM; ++k) x8[k] = xr[k];

    // Layer 1 fused into layer 2 accumulation: h2 stays in registers (64).
    float h2[H2_DIM];
#pragma unroll
    for (int j = 0; j < H2_DIM; ++j) h2[j] = sB2[j];
    for (int i = 0; i < H_DIM; ++i) {
        float t = sB1[i];
#pragma unroll
        for (int k = 0; k < X_DIM; ++k) t = fmaf(x8[k], sW1[k * H_DIM + i], t);
        t = fmaxf(t, 0.0f);
#pragma unroll
        for (int j = 0; j < H2_DIM; ++j) h2[j] = fmaf(t, sW2[i * H2_DIM + j], h2[j]);
    }
#pragma unroll
    for (int j = 0; j < H2_DIM; ++j) h2[j] = fmaxf(h2[j], 0.0f);

    float z[Z_DIM];
#pragma unroll
    for (int e = 0; e < Z_DIM; ++e) z[e] = sB3[e];
    for (int j = 0; j < H2_DIM; ++j) {
#pragma unroll
        for (int e = 0; e < Z_DIM; ++e) z[e] = fmaf(h2[j], sW3[j * Z_DIM + e], z[e]);
    }

    if (is_ctx) {
        float zp[Z_DIM];
#pragma unroll
        for (int e = 0; e < Z_DIM; ++e) zp[e] = 0.0f;
        for (int d = 0; d < Z_DIM; ++d) {
#pragma unroll
            for (int e = 0; e < Z_DIM; ++e) zp[e] = fmaf(z[d], sS[d * Z_DIM + e], zp[e]);
        }
        float c = 0.0f;
#pragma unroll
        for (int e = 0; e < Z_DIM; ++e) c = fmaf(z[e], zp[e], c);
        c *= 0.5f;
#pragma unroll
        for (int e = 0; e < Z_DIM; ++e) zcp_out[(size_t)row * Z_DIM + e] = zp[e];
        c_out[row] = c;
    } else {
        const int tr = row - M_CTX;
#pragma unroll
        for (int e = 0; e < Z_DIM; ++e) zt_out[(size_t)tr * Z_DIM + e] = z[e];
    }
}

// ---------------------------------------------------------------------------
// Stage 2: fused attention.
//   logits L[n,m] = zt[n] . zcp[m] - c[m]; out[n] = softmax_m(L) . y
// Workgroup = 256 threads = 8 waves; each wave owns 16 targets and streams all
// 2048 contexts in 16x16 tiles via V_WMMA_F32_16X16X4_F32 (4 WMMAs per tile,
// K=16). A = zcp tile (contexts = WMMA M rows), B = zt^T (targets = N cols),
// so the online softmax over contexts reduces the 8 accumulator VGPRs
// *within* each lane; the two lane-halves (ctx rows 0-7 / 8-15) merge with a
// single shfl_xor(16) at the end.
// All context features (zcp 128KB + interleaved {c,y} 16KB) are staged once
// into the WGP's 320KB LDS.
// ---------------------------------------------------------------------------
__global__ __launch_bounds__(256) void np_attn_kernel(
    const float* __restrict__ zcp, const float* __restrict__ cbias,
    const float* __restrict__ y_ctx, const float* __restrict__ zt,
    float* __restrict__ out)
{
    __shared__ float sZcp[M_CTX * Z_DIM];   // 128 KB
    __shared__ float sCy[M_CTX * 2];        // 16 KB interleaved {c[m], y[m]}

    const int tid = threadIdx.x;
    {   // cooperative staging: global (L2-resident) -> LDS
        const v4f* src = (const v4f*)zcp;
        v4f*       dst = (v4f*)sZcp;
        for (int i = tid; i < M_CTX * Z_DIM / 4; i += 256) dst[i] = src[i];
        for (int m = tid; m < M_CTX; m += 256) {
            sCy[2 * m]     = cbias[m];
            sCy[2 * m + 1] = y_ctx[m];
        }
    }
    __syncthreads();

    const int lane = tid & 31;
    const int wave = tid >> 5;
    const int r = lane & 15;      // target column within tile (WMMA N)
    const int h = lane >> 4;      // lane-half: K interleave + ctx row group
    const int tgt = blockIdx.x * 128 + wave * 16 + r;

    // B operand (zt^T) is loop-invariant: per lane, 8 of the 16 K values.
    const float* ztr = zt + (size_t)tgt * Z_DIM;
    v2f bz[4];
#pragma unroll
    for (int kc = 0; kc < 4; ++kc)
        bz[kc] = *(const v2f*)(ztr + 4 * kc + 2 * h);

    float mx = -INFINITY, se = 0.0f, acc = 0.0f;

    for (int base = 0; base < M_CTX; base += 16) {
        // A operand: zcp rows (contexts) from LDS, same K interleave as A16x4.
        const float* zr = sZcp + (size_t)(base + r) * Z_DIM;
        v2f a0 = *(const v2f*)(zr + 0  + 2 * h);
        v2f a1 = *(const v2f*)(zr + 4  + 2 * h);
        v2f a2 = *(const v2f*)(zr + 8  + 2 * h);
        v2f a3 = *(const v2f*)(zr + 12 + 2 * h);

        v8f d = {0.f, 0.f, 0.f, 0.f, 0.f, 0.f, 0.f, 0.f};
        d = __builtin_amdgcn_wmma_f32_16x16x4_f32(false, a0, false, bz[0], (short)0, d, false, false);
        d = __builtin_amdgcn_wmma_f32_16x16x4_f32(false, a1, false, bz[1], (short)0, d, false, false);
        d = __builtin_amdgcn_wmma_f32_16x16x4_f32(false, a2, false, bz[2], (short)0, d, false, false);
        d = __builtin_amdgcn_wmma_f32_16x16x4_f32(false, a3, false, bz[3], (short)0, d, false, false);

        // Online softmax over the 8 context rows this lane holds.
        float lg[8], yv[8];
#pragma unroll
        for (int v = 0; v < 8; ++v) {
            const int m = base + v + 8 * h;
            v2f cy = *(const v2f*)(sCy + 2 * m);
            lg[v] = d[v] - cy.x;
            yv[v] = cy.y;
        }
        float tm = lg[0];
#pragma unroll
        for (int v = 1; v < 8; ++v) tm = fmaxf(tm, lg[v]);
        const float nm = fmaxf(mx, tm);
        const float sc = __expf(mx - nm);
        se *= sc;
        acc *= sc;
#pragma unroll
        for (int v = 0; v < 8; ++v) {
            const float e = __expf(lg[v] - nm);
            se += e;
            acc = fmaf(e, yv[v], acc);
        }
        mx = nm;
    }

    // Merge the two lane-halves (ctx rows 0-7 vs 8-15 of every tile).
    const float omx = __shfl_xor(mx, 16, 32);
    const float ose = __shfl_xor(se, 16, 32);
    const float oac = __shfl_xor(acc, 16, 32);
    const float nm  = fmaxf(mx, omx);
    const float s0  = __expf(mx - nm);
    const float s1  = __expf(omx - nm);
    const float den = fmaf(se, s0, ose * s1);
    const float num = fmaf(acc, s0, oac * s1);
    if (h == 0) out[tgt] = num / den;
}

// ---------------------------------------------------------------------------
extern "C" void kernel_launch(void* const* d_in, const int* in_sizes, int n_in,
                              void* d_out, int out_size, void* d_ws, size_t ws_size,
                              hipStream_t stream)
{
    const float* x_context = (const float*)d_in[0];
    const float* y_context = (const float*)d_in[1];
    const float* x_target  = (const float*)d_in[2];
    const float* W1 = (const float*)d_in[3];
    const float* b1 = (const float*)d_in[4];
    const float* W2 = (const float*)d_in[5];
    const float* b2 = (const float*)d_in[6];
    const float* W3 = (const float*)d_in[7];
    const float* b3 = (const float*)d_in[8];
    const float* Wq = (const float*)d_in[9];
    float* out = (float*)d_out;

    // Workspace layout
    char* ws = (char*)d_ws;
    float* zcp = (float*)ws;                                         // 2048*16 f32
    float* cb  = (float*)(ws + (size_t)M_CTX * Z_DIM * 4);           // 2048 f32
    float* ztw = (float*)(ws + (size_t)M_CTX * Z_DIM * 4 + M_CTX*4); // 8192*16 f32

    const int rows = M_CTX + N_TGT;
    np_feat_kernel<<<dim3((rows + 255) / 256), dim3(256), 0, stream>>>(
        x_context, x_target, W1, b1, W2, b2, W3, b3, Wq, zcp, cb, ztw);

    np_attn_kernel<<<dim3(N_TGT / 128), dim3(256), 0, stream>>>(
        zcp, cb, y_context, ztw, out);
}